// VRowSegmentDropout_51496657879217
// MI455X (gfx1250) — compile-verified
//
#include <hip/hip_runtime.h>
#include <stdint.h>

// ---------------------------------------------------------------------------
// Problem constants:  v : (32, 16, 785, 64) f32 ;  out = v * mask(b,h,s,1)
// ---------------------------------------------------------------------------
#define B_   32
#define H_   16
#define S_   785
#define D_   64
#define SIDE_ 28
#define MASK_N   (B_ * H_ * S_)            // 401,920 mask scalars
#define TOTAL_F  ((size_t)MASK_N * D_)     // 25,722,880 floats
#define TOTAL_F4 (TOTAL_F / 4)             // 6,430,720 float4

// TDM streaming geometry: one chunk = 8 mask-rows = 512 floats = 2 KB
#define CHUNK_F  512
#define CHUNK_E8 256                        // 8-byte elements per chunk
#define NCHUNK   (MASK_N / 8)               // 50,240 chunks (exact)
#define WPB      8                          // waves per 256-thread block

typedef float    v4f __attribute__((ext_vector_type(4)));
typedef unsigned v4u __attribute__((ext_vector_type(4)));
typedef int      v8i_t __attribute__((ext_vector_type(8)));
typedef int      v4i_t __attribute__((ext_vector_type(4)));

// ---------------------------------------------------------------------------
// threefry2x32 — exact JAX rotation/injection schedule (20 rounds).
// ---------------------------------------------------------------------------
__device__ __forceinline__ unsigned rotl32(unsigned x, int r) {
    return (x << r) | (x >> (32 - r));
}

__device__ void threefry2x32(unsigned k0, unsigned k1, unsigned c0, unsigned c1,
                             unsigned& o0, unsigned& o1) {
    const unsigned ks2 = k0 ^ k1 ^ 0x1BD11BDAu;
    unsigned x0 = c0 + k0;
    unsigned x1 = c1 + k1;
    const int RA[4] = {13, 15, 26, 6};
    const int RB[4] = {17, 29, 16, 24};
#pragma unroll
    for (int i = 0; i < 4; ++i) { x0 += x1; x1 = rotl32(x1, RA[i]); x1 ^= x0; }
    x0 += k1;  x1 += ks2 + 1u;
#pragma unroll
    for (int i = 0; i < 4; ++i) { x0 += x1; x1 = rotl32(x1, RB[i]); x1 ^= x0; }
    x0 += ks2; x1 += k0 + 2u;
#pragma unroll
    for (int i = 0; i < 4; ++i) { x0 += x1; x1 = rotl32(x1, RA[i]); x1 ^= x0; }
    x0 += k0;  x1 += k1 + 3u;
#pragma unroll
    for (int i = 0; i < 4; ++i) { x0 += x1; x1 = rotl32(x1, RB[i]); x1 ^= x0; }
    x0 += k1;  x1 += ks2 + 4u;
#pragma unroll
    for (int i = 0; i < 4; ++i) { x0 += x1; x1 = rotl32(x1, RA[i]); x1 ^= x0; }
    o0 = x0 + ks2;
    o1 = x1 + k0 + 5u;
}

// Element i of JAX-style random_bits over an even-length iota array of size N:
// counters split into halves [0,N/2) and [N/2,N); element i < N/2 is o0 of
// pair (i, i+N/2), else o1 of pair (i-N/2, i).
__device__ __forceinline__ unsigned rb(unsigned k0, unsigned k1,
                                       unsigned i, unsigned N) {
    const unsigned Hh = N >> 1;
    unsigned o0, o1;
    if (i < Hh) { threefry2x32(k0, k1, i, i + Hh, o0, o1); return o0; }
    threefry2x32(k0, k1, i - Hh, i, o0, o1);
    return o1;
}

// uniform [0,1): (bits>>9 | 1.0f-pattern) as float, minus 1 (JAX mapping)
__device__ __forceinline__ float uf(unsigned bits) {
    return __uint_as_float((bits >> 9) | 0x3F800000u) - 1.0f;
}

// ---------------------------------------------------------------------------
// Kernel 1: generate the (B,H,S) mask into d_ws.  ~0.4M threads, trivial cost.
// ---------------------------------------------------------------------------
__global__ void __launch_bounds__(256)
mask_kernel(float* __restrict__ mask) {
    // per-block precompute of head->dropindex map and per-head subkeys
    __shared__ int      h2j[H_];
    __shared__ unsigned krow[8][2], kst[8][2], kdr[8][2];
    if (threadIdx.x == 0) {
        const unsigned m0 = 0u, m1 = 42u;          // jax.random.key(42)
        // split(master,2) -> k_perm, k_masks
        unsigned kp0 = rb(m0, m1, 0, 4), kp1 = rb(m0, m1, 1, 4);
        unsigned km0 = rb(m0, m1, 2, 4), km1 = rb(m0, m1, 3, 4);
        // permutation of 16 heads (Fisher-Yates on threefry bits), take first 8
        int a[H_];
        for (int i = 0; i < H_; ++i) a[i] = i;
        for (int i = H_ - 1; i > 0; --i) {
            unsigned r  = rb(kp0, kp1, (unsigned)(H_ - 1 - i), 32);
            unsigned sw = (unsigned)(((unsigned long long)r * (unsigned)(i + 1)) >> 32);
            int t = a[i]; a[i] = a[sw]; a[sw] = t;
        }
        for (int i = 0; i < H_; ++i) h2j[i] = -1;
        for (int j = 0; j < 8; ++j)  h2j[a[j]] = j;
        // head_keys = split(k_masks, 8); then split each into (k_row,k_start,k_drop)
        for (int j = 0; j < 8; ++j) {
            unsigned hk0 = rb(km0, km1, 2u * j,     16);
            unsigned hk1 = rb(km0, km1, 2u * j + 1, 16);
            krow[j][0] = rb(hk0, hk1, 0, 6); krow[j][1] = rb(hk0, hk1, 1, 6);
            kst [j][0] = rb(hk0, hk1, 2, 6); kst [j][1] = rb(hk0, hk1, 3, 6);
            kdr [j][0] = rb(hk0, hk1, 4, 6); kdr [j][1] = rb(hk0, hk1, 5, 6);
        }
    }
    __syncthreads();

    unsigned gid = blockIdx.x * blockDim.x + threadIdx.x;
    if (gid >= MASK_N) return;
    unsigned s  = gid % S_;
    unsigned bh = gid / S_;
    unsigned h  = bh % H_;
    unsigned b  = bh / H_;

    float m = 1.0f;
    int j = h2j[h];
    if (j >= 0 && s > 0) {                          // s==0 is the CLS token
        unsigned p   = s - 1;
        unsigned row = p / SIDE_;
        unsigned col = p % SIDE_;
        unsigned idx = b * SIDE_ + row;             // flat (batch,row), N=896
        if (uf(rb(krow[j][0], krow[j][1], idx, 896)) < 0.1f) {     // ROW_P
            unsigned sb    = rb(kst[j][0], kst[j][1], idx, 896);
            unsigned start = (unsigned)(((unsigned long long)sb * 27ull) >> 32); // [0,26]
            int off = (int)col - (int)start;
            if (off >= 0 && off < 2) {                              // BAND_WIDTH
                unsigned db = rb(kdr[j][0], kdr[j][1], idx * 2u + (unsigned)off, 1792);
                if (uf(db) < 0.8f) m = 0.0f;                        // SEG_DROP_P
            }
        }
    }
    mask[gid] = m;
}

// ---------------------------------------------------------------------------
// Kernel 2: out = v * broadcast(mask).  Memory-bound: 206 MB @ 23.3 TB/s.
// Primary path: Tensor Data Mover stages v into LDS (double buffered,
// s_wait_tensorcnt), waves multiply from LDS and NT-store b128 to out.
// Fallback: plain grid-stride float4 NT load/store.
// ---------------------------------------------------------------------------
#if defined(__HIP_DEVICE_COMPILE__) && \
    __has_builtin(__builtin_amdgcn_tensor_load_to_lds)
#define HAVE_TDM 1
#else
#define HAVE_TDM 0
#endif

#if HAVE_TDM
#if __has_builtin(__builtin_amdgcn_s_wait_tensorcnt)
#define TWAIT(n) __builtin_amdgcn_s_wait_tensorcnt((short)(n))
#else
#define TWAIT(n) asm volatile("s_wait_tensorcnt %0" :: "i"(n) : "memory")
#endif

// Build a 1-D tensor descriptor (D#) per CDNA5 ISA ch.8 and issue TDM load:
//   data_size = 8B, tile_dim0 = tensor_dim0 = 256 elems (2 KB), type = 2.
// Toolchain here is the 6-arg form:
//   (uint32x4 g0, int32x8 g1, int32x4 g2, int32x4 g3, int32x8 g4, i32 cpol)
__device__ __forceinline__ void tdm_load(float* ldsDst, const float* gsrc) {
    unsigned lds = (unsigned)(uintptr_t)ldsDst;              // LDS byte offset
    unsigned long long ga = (unsigned long long)(uintptr_t)gsrc;
    v4u g0;
    g0.x = 1u;                                   // count=1, user mode, no gather
    g0.y = lds;                                  // [63:32]  lds_addr
    g0.z = (unsigned)ga;                         // [95:64]  global_addr lo
    g0.w = ((unsigned)(ga >> 32) & 0x01FFFFFFu)  // [120:96] global_addr hi
           | (2u << 30);                         // [127:126] type=2 (image)
    v8i_t g1;
    g1[0] = (int)(3u << 16);                     // data_size=3 (8B); wg_mask=0
    g1[1] = (int)((unsigned)CHUNK_E8 << 16);     // tensor_dim0[15:0]  @ [63:48]
    g1[2] = (int)(1u << 16);                     // tensor_dim0 hi=0; tensor_dim1=1
    g1[3] = (int)((unsigned)CHUNK_E8 << 16);     // tensor_dim1 hi=0; tile_dim0=256
    g1[4] = 0;                                   // tile_dim1=0 (1-D), tile_dim2=0
    g1[5] = CHUNK_E8;                            // tensor_dim0_stride lo
    g1[6] = 0;                                   // stride hi / dim1_stride lo
    g1[7] = 0;
    v4i_t z4 = {0, 0, 0, 0};                     // group2 (dims 2/3 unused)
    v8i_t z8 = {0, 0, 0, 0, 0, 0, 0, 0};         // extended/reserved group
    __builtin_amdgcn_tensor_load_to_lds(g0, g1, z4, z4, z8, 0);
}
#endif  // HAVE_TDM

__global__ void __launch_bounds__(256)
apply_kernel(const float* __restrict__ v, const float* __restrict__ mask,
             float* __restrict__ out) {
#if HAVE_TDM
    __shared__ __align__(16) float smem[WPB][2][CHUNK_F];   // 32 KB / block
    const int lane = threadIdx.x & 31;
    const int wib  = threadIdx.x >> 5;                      // wave in block
    const int wave = blockIdx.x * WPB + wib;
    const int nw   = gridDim.x * WPB;

    int c = wave;
    if (c < NCHUNK) tdm_load(&smem[wib][0][0], v + (size_t)c * CHUNK_F);

    int buf = 0;
    for (; c < NCHUNK; c += nw, buf ^= 1) {
        const int cn = c + nw;
        if (cn < NCHUNK) {
            tdm_load(&smem[wib][buf ^ 1][0], v + (size_t)cn * CHUNK_F);
            TWAIT(1);                       // wait for current buffer only
        } else {
            TWAIT(0);
        }
        asm volatile("" ::: "memory");      // keep LDS reads below the wait
        const float* lb   = &smem[wib][buf][0];
        const size_t base = (size_t)c * CHUNK_F;
#pragma unroll
        for (int p = 0; p < 4; ++p) {
            const int f = p * 128 + lane * 4;           // 128 floats / pass
            v4f x = *(const v4f*)(lb + f);              // ds_load_b128
            const float mv = mask[c * 8 + (f >> 6)];    // one scalar / 64 floats
            x *= mv;
            __builtin_nontemporal_store(x, (v4f*)(out + base + f));
        }
        asm volatile("" ::: "memory");      // reads done before buffer reuse
    }
#else
    // Fallback: grid-stride float4 streaming multiply with NT hints.
    size_t i      = (size_t)blockIdx.x * blockDim.x + threadIdx.x;
    const size_t stride = (size_t)gridDim.x * blockDim.x;
    for (; i < TOTAL_F4; i += stride) {
        v4f x = __builtin_nontemporal_load((const v4f*)v + i);
        x *= mask[i >> 4];
        __builtin_nontemporal_store(x, (v4f*)out + i);
    }
#endif
}

// ---------------------------------------------------------------------------
// Entry point
// ---------------------------------------------------------------------------
extern "C" void kernel_launch(void* const* d_in, const int* in_sizes, int n_in,
                              void* d_out, int out_size, void* d_ws, size_t ws_size,
                              hipStream_t stream) {
    const float* v   = (const float*)d_in[0];
    float*       out = (float*)d_out;
    float*       msk = (float*)d_ws;            // 401,920 floats = 1.6 MB scratch

    (void)in_sizes; (void)n_in; (void)out_size; (void)ws_size;

    mask_kernel<<<(MASK_N + 255) / 256, 256, 0, stream>>>(msk);
    apply_kernel<<<2048, 256, 0, stream>>>(v, msk, out);
}